// Attention_45792941310497
// MI455X (gfx1250) — compile-verified
//
#include <hip/hip_runtime.h>
#include <hip/hip_bf16.h>
#include <math.h>

typedef __attribute__((ext_vector_type(2))) float v2f;
typedef __attribute__((ext_vector_type(8))) float v8f;

#define BB     512
#define TT     128
#define DIN    512
#define HID    256
#define CLS    96
#define NSTEP  26          // batch_max_length + 1
#define GRU_IN (DIN + CLS) // 608

#define WMMA_F32(a, b, c) \
    __builtin_amdgcn_wmma_f32_16x16x4_f32(false, (a), false, (b), (short)0, (c), false, false)

// ---------------------------------------------------------------------------
// fp32 WMMA GEMM:  C[M,N] = A[M,K] @ W[N,K]^T (+ bias[N])
// One wave computes a 32x32 output region = 2x2 tiles of 16x16 via
// V_WMMA_F32_16X16X4_F32, with 2x2 register blocking: each k-step does
// 4 fragment loads feeding 4 WMMAs (1.0 load/WMMA instead of 2.0),
// and 4 independent accumulator chains for pipelining.
// Fragment layouts (ISA 7.12.2):
//  A 16x4 f32: lanes 0-15 hold K={k,k+1} of row M=lane; lanes 16-31 K={k+2,k+3}
//  B  4x16:    symmetric, sourced from rows of W (W row-major [N,K])
//  C/D 16x16:  element v of lane -> M = v + 8*(lane>=16), N = lane&15
// M and N are multiples of 32 for every call site; K multiples of 8.
// ---------------------------------------------------------------------------
__global__ __launch_bounds__(256) void gemm_wmma_f32(
    const float* __restrict__ A, const float* __restrict__ W,
    const float* __restrict__ bias, float* __restrict__ C,
    int M, int N, int K)
{
    const int wave = (int)((blockIdx.x * blockDim.x + threadIdx.x) >> 5);
    const int lane = (int)(threadIdx.x & 31);
    const int nt32 = N >> 5;                 // # of 32-wide column blocks
    const int mt   = wave / nt32;            // 32-high row block
    const int nt   = wave - mt * nt32;
    if ((mt << 5) >= M) return;              // wave-uniform: EXEC all-1s for WMMA

    const int half = lane >> 4;              // 0: lanes 0-15, 1: lanes 16-31
    const int lr   = lane & 15;

    const float* __restrict__ A0 = A + (size_t)((mt << 5) + lr) * K + 2 * half;
    const float* __restrict__ A1 = A0 + (size_t)16 * K;
    const float* __restrict__ W0 = W + (size_t)((nt << 5) + lr) * K + 2 * half;
    const float* __restrict__ W1 = W0 + (size_t)16 * K;

    v8f acc00 = {}, acc01 = {}, acc10 = {}, acc11 = {};
#pragma unroll 2
    for (int k = 0; k < K; k += 4) {
        v2f a0 = *reinterpret_cast<const v2f*>(A0 + k);
        v2f a1 = *reinterpret_cast<const v2f*>(A1 + k);
        v2f b0 = *reinterpret_cast<const v2f*>(W0 + k);
        v2f b1 = *reinterpret_cast<const v2f*>(W1 + k);
        acc00 = WMMA_F32(a0, b0, acc00);
        acc01 = WMMA_F32(a0, b1, acc01);
        acc10 = WMMA_F32(a1, b0, acc10);
        acc11 = WMMA_F32(a1, b1, acc11);
    }

    const float bv0 = bias ? bias[(nt << 5) + lr]      : 0.0f;
    const float bv1 = bias ? bias[(nt << 5) + 16 + lr] : 0.0f;

    float* __restrict__ Cr0 = C + (size_t)((mt << 5) + 8 * half) * N + (nt << 5) + lr;
    float* __restrict__ Cr1 = Cr0 + (size_t)16 * N;
#pragma unroll
    for (int v = 0; v < 8; ++v) {
        Cr0[(size_t)v * N]      = acc00[v] + bv0;
        Cr0[(size_t)v * N + 16] = acc01[v] + bv1;
        Cr1[(size_t)v * N]      = acc10[v] + bv0;
        Cr1[(size_t)v * N + 16] = acc11[v] + bv1;
    }
}

// ---------------------------------------------------------------------------
// Per-step attention: scores -> softmax -> context; writes x = [context|onehot]
// One block (256 thr = 8 waves, wave32) per batch row.
// ---------------------------------------------------------------------------
__device__ __forceinline__ float wave_sum(float p) {
#pragma unroll
    for (int off = 16; off; off >>= 1) p += __shfl_down(p, off, 32);
    return p;
}

__global__ __launch_bounds__(256) void attn_context(
    const float* __restrict__ Hproj,   // [B,T,H]  (i2h projection)
    const float* __restrict__ hp,      // [B,H]    (h2h proj + b_h2h)
    const float* __restrict__ Wscore,  // [H]
    const float* __restrict__ batchH,  // [B,T,D]
    const int*   __restrict__ text,    // [B,NSTEP]
    int step,
    float* __restrict__ x)             // [B, D+C]
{
    const int b    = blockIdx.x;
    const int tid  = (int)threadIdx.x;
    const int wave = tid >> 5, lane = tid & 31;

    __shared__ float s_hp[HID];
    __shared__ float s_e[TT];

    s_hp[tid] = hp[(size_t)b * HID + tid];
    __syncthreads();

    // scores: each wave owns 16 timesteps, lanes parallel over H (coalesced)
    for (int tt = 0; tt < 16; ++tt) {
        const int t = wave * 16 + tt;
        const float* __restrict__ row = Hproj + ((size_t)b * TT + t) * HID;
        float p = 0.0f;
#pragma unroll
        for (int h = lane; h < HID; h += 32)
            p += Wscore[h] * tanhf(row[h] + s_hp[h]);
        p = wave_sum(p);
        if (lane == 0) s_e[t] = p;
    }
    __syncthreads();

    // softmax over T=128 (tiny: every thread scans serially in registers)
    float m = -3.402823466e+38f;
    for (int t = 0; t < TT; ++t) m = fmaxf(m, s_e[t]);
    float ssum = 0.0f;
    for (int t = 0; t < TT; ++t) ssum += expf(s_e[t] - m);
    const float inv = 1.0f / ssum;
    float my_a = (tid < TT) ? expf(s_e[tid] - m) * inv : 0.0f;
    __syncthreads();
    if (tid < TT) s_e[tid] = my_a;
    __syncthreads();

    // context[d] = sum_t alpha[t] * batch_H[b,t,d]  (lanes over d: coalesced)
    for (int d = tid; d < DIN; d += 256) {
        const float* __restrict__ Hb = batchH + ((size_t)b * TT) * DIN + d;
        float c = 0.0f;
        for (int t = 0; t < TT; ++t) c += s_e[t] * Hb[(size_t)t * DIN];
        x[(size_t)b * GRU_IN + d] = c;
    }
    // one-hot(char) tail of x
    if (tid < CLS) {
        const int ch = text[(size_t)b * NSTEP + step];
        x[(size_t)b * GRU_IN + DIN + tid] = (tid == ch) ? 1.0f : 0.0f;
    }
}

// ---------------------------------------------------------------------------
// GRU gate combine (torch semantics), in-place hidden update + history write
// ---------------------------------------------------------------------------
__device__ __forceinline__ float sigmoidf(float v) { return 1.0f / (1.0f + expf(-v)); }

__global__ __launch_bounds__(256) void gru_gate(
    const float* __restrict__ gi,      // [B,3H]
    const float* __restrict__ gh,      // [B,3H]
    float* __restrict__ hidden,        // [B,H] read old, write new
    float* __restrict__ hiddens,       // [B,NSTEP,H]
    int step)
{
    const int i = (int)(blockIdx.x * blockDim.x + threadIdx.x); // 0..B*H
    const int b = i >> 8;        // HID == 256
    const int h = i & (HID - 1);
    const float* __restrict__ gib = gi + (size_t)b * 3 * HID;
    const float* __restrict__ ghb = gh + (size_t)b * 3 * HID;

    const float r  = sigmoidf(gib[h]           + ghb[h]);
    const float z  = sigmoidf(gib[HID + h]     + ghb[HID + h]);
    const float n  = tanhf  (gib[2 * HID + h] + r * ghb[2 * HID + h]);
    const float hv = hidden[i];
    const float nh = (1.0f - z) * n + z * hv;
    hidden[i] = nh;
    hiddens[((size_t)b * NSTEP + step) * HID + h] = nh;
}

// ---------------------------------------------------------------------------
extern "C" void kernel_launch(void* const* d_in, const int* in_sizes, int n_in,
                              void* d_out, int out_size, void* d_ws, size_t ws_size,
                              hipStream_t stream)
{
    (void)in_sizes; (void)n_in; (void)out_size; (void)ws_size;

    const float* batch_H = (const float*)d_in[0];
    const int*   text    = (const int*)  d_in[1];
    /* d_in[2] = batch_max_length scalar; fixed at 25 -> NSTEP=26 */
    const float* W_i2h   = (const float*)d_in[3];
    const float* W_h2h   = (const float*)d_in[4];
    const float* b_h2h   = (const float*)d_in[5];
    const float* W_score = (const float*)d_in[6];
    const float* W_ih    = (const float*)d_in[7];
    const float* W_hh    = (const float*)d_in[8];
    const float* b_ih    = (const float*)d_in[9];
    const float* b_hh    = (const float*)d_in[10];
    const float* W_gen   = (const float*)d_in[11];
    const float* b_gen   = (const float*)d_in[12];
    float* out = (float*)d_out;

    // workspace carve-out (256B aligned), ~86 MB total
    char* ws = (char*)d_ws;
    size_t off = 0;
    auto alloc = [&](size_t nf) {
        float* p = (float*)(ws + off);
        off += ((nf * sizeof(float) + 255) / 256) * 256;
        return p;
    };
    float* Hproj   = alloc((size_t)BB * TT * HID);   // 64 MB
    float* xbuf    = alloc((size_t)BB * GRU_IN);
    float* hp      = alloc((size_t)BB * HID);
    float* gi      = alloc((size_t)BB * 3 * HID);
    float* gh      = alloc((size_t)BB * 3 * HID);
    float* hidden  = alloc((size_t)BB * HID);
    float* hiddens = alloc((size_t)BB * NSTEP * HID);

    hipMemsetAsync(hidden, 0, (size_t)BB * HID * sizeof(float), stream); // h0 = 0

    auto gemm = [&](const float* A, const float* W, const float* bias,
                    float* C, int M, int N, int K) {
        const int tiles  = (M / 32) * (N / 32);  // 32x32 region per wave
        const int blocks = (tiles + 7) / 8;      // 8 waves per block
        gemm_wmma_f32<<<blocks, 256, 0, stream>>>(A, W, bias, C, M, N, K);
    };

    // loop-invariant i2h projection: [B*T, H]
    gemm(batch_H, W_i2h, nullptr, Hproj, BB * TT, HID, DIN);

    for (int s = 0; s < NSTEP; ++s) {
        gemm(hidden, W_h2h, b_h2h, hp, BB, HID, HID);                    // h2h proj
        attn_context<<<BB, 256, 0, stream>>>(Hproj, hp, W_score, batch_H,
                                             text, s, xbuf);             // attn+ctx
        gemm(xbuf,   W_ih, b_ih, gi, BB, 3 * HID, GRU_IN);               // x @ W_ih^T
        gemm(hidden, W_hh, b_hh, gh, BB, 3 * HID, HID);                  // h @ W_hh^T
        gru_gate<<<(BB * HID) / 256, 256, 0, stream>>>(gi, gh, hidden,
                                                       hiddens, s);      // gates
    }

    // generator: probs[B*NSTEP, C] = hiddens @ W_gen^T + b_gen
    gemm(hiddens, W_gen, b_gen, out, BB * NSTEP, CLS, HID);
}